// Encoder_64819646431326
// MI455X (gfx1250) — compile-verified
//
#include <hip/hip_runtime.h>

// DA-RNN input-attention encoder for MI455X (gfx1250, wave32).
// B=128, T=256, N=256, H=256.
//
// Structure:
//  1) uproj_wmma_kernel: the only large dense GEMM (u_proj = inputs_scan @ Wu,
//     32768x256 @ 256x256, fp32) -> V_WMMA_F32_16X16X4_F32, 1 tile/wave.
//  2) scan_kernel: persistent kernel, ONE workgroup per batch element runs the
//     whole T=256 recurrence (the recurrence is independent across b, so no
//     grid sync / kernel relaunch chain is needed). All state in LDS; u_proj
//     (33.5 MB) + inputs (33.5 MB) + weights stay resident in the 192 MB L2,
//     so the scan is L2-/TRANS-bound, not HBM-bound.

#define B_ 128
#define T_ 256
#define N_ 256
#define H_ 256

typedef __attribute__((ext_vector_type(2))) float v2f;
typedef __attribute__((ext_vector_type(8))) float v8f;

__device__ __forceinline__ float fast_tanh(float x) {
#if __has_builtin(__builtin_amdgcn_tanhf)
  return __builtin_amdgcn_tanhf(x);            // v_tanh_f32 (gfx1250 TRANS op)
#else
  float e = __expf(x + x);                     // v_exp_f32
  return 1.f - 2.f * __builtin_amdgcn_rcpf(e + 1.f);
#endif
}

__device__ __forceinline__ float fast_sigmoid(float x) {
  return __builtin_amdgcn_rcpf(1.f + __expf(-x));
}

// ---------------------------------------------------------------------------
// Kernel 1: u_proj[m= b*N+n][t'] = sum_k inputs[b,k,n] * Wu[k,t'] + bu[t']
// fp32 WMMA 16x16x4. A-frag layout (ISA 7.12.2): lane<16 holds M=lane,
// K={0,1} in v[0],v[1]; lane>=16 holds M=lane-16, K={2,3}. B-frag mirrored
// (N=lane&15, K split across lane halves). D: VGPR r -> M = r + 8*(lane>>4),
// N = lane&15.
// ---------------------------------------------------------------------------
__global__ __launch_bounds__(256) void uproj_wmma_kernel(
    const float* __restrict__ inp,   // [B,T,N]
    const float* __restrict__ Wu,    // [T,T]
    const float* __restrict__ bu,    // [T]
    float* __restrict__ uproj)       // [B,N,T]
{
  const int lane  = threadIdx.x & 31;
  const int wave  = threadIdx.x >> 5;
  const int j0    = blockIdx.x * 16;                  // column tile (t')
  const int tileM = (blockIdx.y * 8 + wave) * 16;     // row tile (m = b*N+n)

  const int half = lane >> 4;       // 0 -> K pair {0,1}, 1 -> K pair {2,3}
  const int mrow = lane & 15;

  const int mA  = tileM + mrow;     // A row for this lane
  const int bA  = mA >> 8;          // N_ == 256
  const int nA  = mA & 255;
  const int col = j0 + mrow;        // B-frag / D column for this lane

  const float bias = bu[col];
  v8f acc = {};

  for (int k = 0; k < T_; k += 4) {
    const int kb = k + half * 2;
    v2f a, b;
    // A: inputs_scan[b, n, k] = inputs[b, k, n] (lanes contiguous in n)
    a.x = inp[((size_t)bA * T_ + kb    ) * N_ + nA];
    a.y = inp[((size_t)bA * T_ + kb + 1) * N_ + nA];
    // B: Wu[k, col] (lanes contiguous in col)
    b.x = Wu[(size_t)(kb    ) * T_ + col];
    b.y = Wu[(size_t)(kb + 1) * T_ + col];
    acc = __builtin_amdgcn_wmma_f32_16x16x4_f32(
        /*neg_a=*/false, a, /*neg_b=*/false, b,
        /*c_mod=*/(short)0, acc, /*reuse_a=*/false, /*reuse_b=*/false);
  }

#pragma unroll
  for (int r = 0; r < 8; ++r) {
    const int mD = tileM + r + 8 * half;
    uproj[(size_t)mD * T_ + col] = acc[r] + bias;
  }
}

// ---------------------------------------------------------------------------
// Kernel 2: persistent scan. grid = B blocks, 256 threads (8 wave32s).
// Each block owns batch element b for all 256 steps; h,c,s,weights-vectors
// live in LDS; no inter-block communication is ever required.
// ---------------------------------------------------------------------------
__global__ __launch_bounds__(256) void scan_kernel(
    const float* __restrict__ inp,    // [B,T,N]
    const float* __restrict__ h0,     // [B,H]
    const float* __restrict__ c0,     // [B,H]
    const float* __restrict__ Ww,     // [2H,T]
    const float* __restrict__ bw,     // [T]
    const float* __restrict__ Wv,     // [T,1]
    const float* __restrict__ bvp,    // [1]
    const float* __restrict__ Wfc,    // [H+N,H]
    const float* __restrict__ bfc,    // [H]
    const float* __restrict__ uproj,  // [B,N,T]
    float* __restrict__ out)          // [B,T,H]
{
  __shared__ float hS[H_], cS[H_], sS[T_], scS[N_], xwS[N_];
  __shared__ float bwS[T_], bfcS[H_], wvS[T_];
  __shared__ float red[8];

  const int tid  = threadIdx.x;
  const int b    = blockIdx.x;
  const int lane = tid & 31;
  const int wv   = tid >> 5;

  hS[tid]   = h0[b * H_ + tid];
  cS[tid]   = c0[b * H_ + tid];
  bwS[tid]  = bw[tid];
  bfcS[tid] = bfc[tid];
  wvS[tid]  = Wv[tid];
  const float bv = bvp[0];
  __syncthreads();

  const float4* up4base = (const float4*)(uproj + (size_t)b * N_ * T_);

  for (int t = 0; t < T_; ++t) {
    // ---- Phase 1: s[j] = [h;c] . Ww[:,j] + bw[j]  (thread j = tid) --------
    float sacc = bwS[tid];
#pragma unroll 8
    for (int i = 0; i < H_; ++i) sacc += hS[i] * Ww[(size_t)i * T_ + tid];
#pragma unroll 8
    for (int i = 0; i < H_; ++i) sacc += cS[i] * Ww[(size_t)(H_ + i) * T_ + tid];
    sS[tid] = sacc;
    __syncthreads();

    // ---- Phase 2: score[n] = Wv . tanh(u_proj[b,n,:] + s) + bv ------------
    // wave wv handles n = wv*32 .. wv*32+31; lanes split the T reduction
    // (float4 -> 512B coalesced bursts from L2-resident u_proj).
    for (int nsub = 0; nsub < 32; ++nsub) {
      const int n = wv * 32 + nsub;
      const float4* up4 = up4base + ((size_t)n * T_ >> 2);
      float acc = 0.f;
#pragma unroll
      for (int it = 0; it < 2; ++it) {
        const int tt = it * 128 + lane * 4;
        float4 u = up4[tt >> 2];
        acc += wvS[tt + 0] * fast_tanh(u.x + sS[tt + 0]);
        acc += wvS[tt + 1] * fast_tanh(u.y + sS[tt + 1]);
        acc += wvS[tt + 2] * fast_tanh(u.z + sS[tt + 2]);
        acc += wvS[tt + 3] * fast_tanh(u.w + sS[tt + 3]);
      }
      for (int o = 16; o > 0; o >>= 1) acc += __shfl_xor(acc, o, 32);
      if (lane == 0) scS[n] = acc + bv;
    }
    __syncthreads();

    // ---- Phase 3: softmax over n, xw[n] = weight[n] * inputs[b,t,n] -------
    const float sval = scS[tid];
    float m = sval;
    for (int o = 16; o > 0; o >>= 1) m = fmaxf(m, __shfl_xor(m, o, 32));
    if (lane == 0) red[wv] = m;
    __syncthreads();
    if (tid < 8) {
      float x = red[tid];
      for (int o = 4; o > 0; o >>= 1) x = fmaxf(x, __shfl_xor(x, o, 8));
      if (tid == 0) red[0] = x;
    }
    __syncthreads();
    m = red[0];
    const float e = __expf(sval - m);
    __syncthreads();                       // red[] about to be reused
    float ssum = e;
    for (int o = 16; o > 0; o >>= 1) ssum += __shfl_xor(ssum, o, 32);
    if (lane == 0) red[wv] = ssum;
    __syncthreads();
    if (tid < 8) {
      float x = red[tid];
      for (int o = 4; o > 0; o >>= 1) x += __shfl_xor(x, o, 8);
      if (tid == 0) red[0] = x;
    }
    __syncthreads();
    const float wgt = e / red[0];
    xwS[tid] = wgt * inp[((size_t)b * T_ + t) * N_ + tid];
    __syncthreads();

    // ---- Phase 4: g[j] = [h; xw] . Wfc[:,j] + bfc[j] ----------------------
    float g = bfcS[tid];
#pragma unroll 8
    for (int i = 0; i < H_; ++i) g += hS[i] * Wfc[(size_t)i * H_ + tid];
#pragma unroll 8
    for (int i = 0; i < N_; ++i) g += xwS[i] * Wfc[(size_t)(H_ + i) * H_ + tid];

    // ---- Phase 5: shared-gate LSTM update (faithful to reference) ---------
    const float sg = fast_sigmoid(g);
    const float cn = sg * (cS[tid] + fast_tanh(g));   // sg*c + sg*tanh(g)
    const float hn = sg * fast_tanh(cn);
    __syncthreads();                       // all phase-4 reads of hS done
    hS[tid] = hn;
    cS[tid] = cn;
    out[((size_t)b * T_ + t) * H_ + tid] = hn;
    __syncthreads();
  }
}

// ---------------------------------------------------------------------------
extern "C" void kernel_launch(void* const* d_in, const int* in_sizes, int n_in,
                              void* d_out, int out_size, void* d_ws, size_t ws_size,
                              hipStream_t stream) {
  const float* inp = (const float*)d_in[0];   // [B,T,N]
  const float* h0  = (const float*)d_in[1];   // [B,H]
  const float* c0  = (const float*)d_in[2];   // [B,H]
  const float* Ww  = (const float*)d_in[3];   // [2H,T]
  const float* bw  = (const float*)d_in[4];   // [T]
  const float* Wu  = (const float*)d_in[5];   // [T,T]
  const float* bu  = (const float*)d_in[6];   // [T]
  const float* Wv  = (const float*)d_in[7];   // [T,1]
  const float* bv  = (const float*)d_in[8];   // [1]
  const float* Wfc = (const float*)d_in[9];   // [H+N,H]
  const float* bfc = (const float*)d_in[10];  // [H]
  float* out   = (float*)d_out;               // [B,T,H]
  float* uproj = (float*)d_ws;                // [B,N,T] = 32 MB scratch

  // 1) hoisted GEMM via fp32 WMMA: 16 col-tiles x 256 row-tile groups (8 waves)
  dim3 g1(T_ / 16, (B_ * N_) / (16 * 8));
  uproj_wmma_kernel<<<g1, dim3(256), 0, stream>>>(inp, Wu, bu, uproj);

  // 2) persistent per-batch scan: 128 workgroups, zero grid-wide sync
  scan_kernel<<<dim3(B_), dim3(256), 0, stream>>>(inp, h0, c0, Ww, bw, Wv, bv,
                                                  Wfc, bfc, uproj, out);
}